// GNNLayer_7473243095119
// MI455X (gfx1250) — compile-verified
//
#include <hip/hip_runtime.h>

#define Bn 4
#define Vn 256
#define Hd 128
#define BV  (Bn * Vn)        // 1024
#define BVH (BV * Hd)        // 131072
#define BVV (Bn * Vn * Vn)   // 262144

typedef __attribute__((ext_vector_type(2))) float v2f;
typedef __attribute__((ext_vector_type(8))) float v8f;
typedef __attribute__((ext_vector_type(4))) unsigned int u32x4;
typedef __attribute__((ext_vector_type(8))) int i32x8;
typedef __attribute__((ext_vector_type(4))) int i32x4;

#if __has_builtin(__builtin_amdgcn_tensor_load_to_lds) && __has_builtin(__builtin_amdgcn_s_wait_tensorcnt)
#define USE_TDM 1
#else
#define USE_TDM 0
#endif

__device__ __forceinline__ v8f wmma_f32x4(v2f a, v2f b, v8f c) {
  // D(16x16) = A(16x4) * B(4x16) + C, full fp32
  return __builtin_amdgcn_wmma_f32_16x16x4_f32(false, a, false, b, (short)0, c, false, false);
}

// ---------------- K0: zero BN-e accumulators ----------------
__global__ void k_zero(float* p) {
  if (threadIdx.x < 256) p[threadIdx.x] = 0.0f;
}

// ---------------- K1: four h-linears (Uh, Vh, Ah, Bh) via WMMA ----------------
__global__ __launch_bounds__(256) void k_hlin(
    const float* __restrict__ h,
    const float* __restrict__ Uw, const float* __restrict__ Ub,
    const float* __restrict__ Vw, const float* __restrict__ Vb,
    const float* __restrict__ Aw, const float* __restrict__ Ab,
    const float* __restrict__ Bw, const float* __restrict__ Bb,
    float* __restrict__ Uh, float* __restrict__ Vh,
    float* __restrict__ Ah, float* __restrict__ Bh) {
  const int tid  = threadIdx.x;
  const int wave = tid >> 5;
  const int lane = tid & 31;
  const int half = lane >> 4;
  const int l    = lane & 15;

  const int gid  = blockIdx.x * 8 + wave;   // 0..2047
  const int w_id = gid >> 9;                // 512 tiles per weight (64 m-tiles * 8 n-tiles)
  const int rem  = gid & 511;
  const int m0   = (rem >> 3) * 16;
  const int n0   = (rem & 7) * 16;

  const float* W; const float* bs; float* out;
  if      (w_id == 0) { W = Uw; bs = Ub; out = Uh; }
  else if (w_id == 1) { W = Vw; bs = Vb; out = Vh; }
  else if (w_id == 2) { W = Aw; bs = Ab; out = Ah; }
  else                { W = Bw; bs = Bb; out = Bh; }

  v8f acc = {};
  #pragma unroll
  for (int t = 0; t < 32; ++t) {
    const int k0 = 4 * t;
    v2f a, b;
    const float* ap = h + (m0 + l) * Hd + k0 + 2 * half;
    a.x = ap[0]; a.y = ap[1];
    // Bmat[k][n] = W[n][k]  (x @ W.T)
    const float* wp = W + (n0 + l) * Hd + k0 + 2 * half;
    b.x = wp[0]; b.y = wp[1];
    acc = wmma_f32x4(a, b, acc);
  }
  const float bias = bs[n0 + l];
  #pragma unroll
  for (int r = 0; r < 8; ++r) {
    const int m = m0 + r + 8 * half;
    out[m * Hd + n0 + l] = acc[r] + bias;
  }
}

// ---------------- K2: edge GEMM + broadcast + sigmoid-gated agg + BN-e stats ----------------
// TDM double-buffers the 16x128 e tile into LDS with hardware row padding (stride 132 dwords).
__global__ __launch_bounds__(256) void k_emain(
    const float* __restrict__ e,
    const float* __restrict__ Cw, const float* __restrict__ Cb,
    const float* __restrict__ Ah, const float* __restrict__ Bh,
    const float* __restrict__ Vh,
    float* __restrict__ e_new,      // d_out e-region
    float* __restrict__ agg,
    float* __restrict__ sum_e, float* __restrict__ sumsq_e) {
  __shared__ float sE[2][16 * 132];   // padded stride breaks LDS bank conflicts on A reads
  __shared__ float sVh[16 * Hd];
  __shared__ float sAh[16 * Hd];

  const int tid  = threadIdx.x;
  const int wave = tid >> 5;
  const int lane = tid & 31;
  const int half = lane >> 4;
  const int l    = lane & 15;

  const int b = blockIdx.x >> 8;    // / Vn
  const int i = blockIdx.x & 255;
  const int n0 = wave * 16;         // 8 waves cover 128 columns

  // Register-cache B fragments of Cw for this wave's fixed n-tile (reused over 16 j-tiles)
  v2f bf[32];
  #pragma unroll
  for (int t = 0; t < 32; ++t) {
    const float* wp = Cw + (n0 + l) * Hd + 4 * t + 2 * half;
    bf[t].x = wp[0]; bf[t].y = wp[1];
  }
  const float bh_i = Bh[(b * Vn + i) * Hd + n0 + l];
  const float cb_n = Cb[n0 + l];

  const float* eb = e    + (size_t)(b * Vn + i) * Vn * Hd;
  float*       eo = e_new+ (size_t)(b * Vn + i) * Vn * Hd;
  const float* vb = Vh   + (size_t)b * Vn * Hd;
  const float* ab = Ah   + (size_t)b * Vn * Hd;

  float aggp = 0.f, se = 0.f, sq = 0.f;

#if USE_TDM
  // Issue a TDM 2D-tile load: 16 rows x 128 dwords, LDS pad 4 dwords every 128 dwords.
  auto tdm_issue = [&](int p) {
    const unsigned lds = (unsigned)(uintptr_t)(void*)(&sE[p & 1][0]);
    const unsigned long long ga = (unsigned long long)(uintptr_t)(eb + p * 16 * Hd);
    u32x4 g0;
    g0.x = 1u;                                   // count=1 (valid descriptor)
    g0.y = lds;                                  // lds_addr
    g0.z = (unsigned)ga;                         // global_addr[31:0]
    g0.w = ((unsigned)(ga >> 32) & 0x01FFFFFFu) | (2u << 30);  // global_addr[56:32] | type=2
    i32x8 g1;
    g1[0] = (int)((2u << 16)      // data_size = 4 bytes
                | (1u << 20)      // pad_enable
                | (6u << 22)      // pad_interval: 128 dwords
                | (3u << 25));    // pad_amount: 4 dwords
    g1[1] = (int)(128u << 16);    // tensor_dim0[15:0]=128 (atomic_barrier_addr=0)
    g1[2] = (int)(16u << 16);     // tensor_dim0[31:16]=0 | tensor_dim1[15:0]=16
    g1[3] = (int)(128u << 16);    // tensor_dim1[31:16]=0 | tile_dim0=128
    g1[4] = (int)(16u);           // tile_dim1=16 | tile_dim2=0
    g1[5] = (int)(128u);          // tensor_dim0_stride[31:0]=128
    g1[6] = 0;
    g1[7] = 0;
    i32x4 gz4 = {0, 0, 0, 0};
    i32x8 gz8 = {0, 0, 0, 0, 0, 0, 0, 0};
    __builtin_amdgcn_tensor_load_to_lds(g0, g1, gz4, gz4, gz8, 0);
  };
  if (wave == 0) tdm_issue(0);
#endif

  for (int p = 0; p < 16; ++p) {
    const int j0 = p * 16;
    __syncthreads();   // previous tile's compute done: safe to overwrite sVh/sAh & other sE buffer
    // stage Vh & Ah j-tiles (contiguous in memory)
    {
      const float4* v4 = (const float4*)(vb + j0 * Hd);
      const float4* a4 = (const float4*)(ab + j0 * Hd);
      float4* sv4 = (float4*)sVh; float4* sa4 = (float4*)sAh;
      for (int t = tid; t < 512; t += 256) { sv4[t] = v4[t]; sa4[t] = a4[t]; }
    }
#if USE_TDM
    if (wave == 0) {
      if (p + 1 < 16) {
        tdm_issue(p + 1);                          // overlap next DMA with this tile's WMMAs
        __builtin_amdgcn_s_wait_tensorcnt((short)1); // oldest (tile p) complete
      } else {
        __builtin_amdgcn_s_wait_tensorcnt((short)0);
      }
    }
    const float* sEb = &sE[p & 1][0];
#else
    for (int t = tid; t < 2048; t += 256) {
      const int jj = t >> 7, k = t & 127;
      sE[0][jj * 132 + k] = eb[j0 * Hd + t];
    }
    const float* sEb = &sE[0][0];
#endif
    __syncthreads();

    v8f acc = {};
    #pragma unroll
    for (int t = 0; t < 32; ++t) {
      const int k0 = 4 * t;
      v2f a;
      a.x = sEb[l * 132 + k0 + 2 * half];
      a.y = sEb[l * 132 + k0 + 2 * half + 1];
      acc = wmma_f32x4(a, bf[t], acc);
    }
    #pragma unroll
    for (int r = 0; r < 8; ++r) {
      const int jj = r + 8 * half;           // row of tile = neighbor j
      const float val = acc[r] + sAh[jj * Hd + n0 + l] + bh_i + cb_n;
      eo[(j0 + jj) * Hd + n0 + l] = val;     // e_new
      const float g = 1.0f / (1.0f + __expf(-val));
      aggp += g * sVh[jj * Hd + n0 + l];
      se += val; sq += val * val;
    }
  }

  // lanes l and l+16 hold the same column n0+l -> combine
  aggp += __shfl_xor(aggp, 16);
  se   += __shfl_xor(se, 16);
  sq   += __shfl_xor(sq, 16);
  if (lane < 16) {
    agg[(size_t)(b * Vn + i) * Hd + n0 + l] = aggp;
    atomicAdd(&sum_e[n0 + l], se);
    atomicAdd(&sumsq_e[n0 + l], sq);
  }
}

// ---------------- K3: h_new = Uh + agg, per-column BN-h stats (block per column) ----------------
__global__ __launch_bounds__(256) void k_hstats(
    const float* __restrict__ Uh, const float* __restrict__ agg,
    float* __restrict__ h_new, float* __restrict__ sum_h, float* __restrict__ sumsq_h) {
  __shared__ float rs[256], rq[256];
  const int n = blockIdx.x;          // column 0..127
  const int tid = threadIdx.x;
  float s = 0.f, q = 0.f;
  for (int r = tid; r < BV; r += 256) {
    const float v = Uh[r * Hd + n] + agg[r * Hd + n];
    h_new[r * Hd + n] = v;
    s += v; q += v * v;
  }
  rs[tid] = s; rq[tid] = q;
  __syncthreads();
  for (int st = 128; st > 0; st >>= 1) {
    if (tid < st) { rs[tid] += rs[tid + st]; rq[tid] += rq[tid + st]; }
    __syncthreads();
  }
  if (tid == 0) { sum_h[n] = rs[0]; sumsq_h[n] = rq[0]; }
}

// ---------------- K4: stats -> (scale, shift) for both BNs ----------------
__global__ void k_fin(
    const float* __restrict__ sum_e, const float* __restrict__ sumsq_e,
    const float* __restrict__ sum_h, const float* __restrict__ sumsq_h,
    const float* __restrict__ gamma_h, const float* __restrict__ beta_h,
    const float* __restrict__ gamma_e, const float* __restrict__ beta_e,
    float* __restrict__ sc_e, float* __restrict__ sh_e,
    float* __restrict__ sc_h, float* __restrict__ sh_h) {
  const int n = threadIdx.x;
  if (n >= Hd) return;
  {
    const float inv = 1.0f / (float)BVV;
    const float m = sum_e[n] * inv;
    const float v = sumsq_e[n] * inv - m * m;
    const float s = gamma_e[n] * rsqrtf(v + 1e-5f);
    sc_e[n] = s; sh_e[n] = beta_e[n] - m * s;
  }
  {
    const float inv = 1.0f / (float)BV;
    const float m = sum_h[n] * inv;
    const float v = sumsq_h[n] * inv - m * m;
    const float s = gamma_h[n] * rsqrtf(v + 1e-5f);
    sc_h[n] = s; sh_h[n] = beta_h[n] - m * s;
  }
}

// ---------------- K5: h_out = h_in + relu(bn(h_new)) ----------------
__global__ __launch_bounds__(256) void k_happly(
    const float* __restrict__ h_new, const float* __restrict__ sc, const float* __restrict__ sh,
    const float* __restrict__ h_in, float* __restrict__ h_out) {
  const int t = blockIdx.x * 256 + threadIdx.x;
  const int n = t & 127;
  const float y = h_new[t] * sc[n] + sh[n];
  h_out[t] = h_in[t] + fmaxf(y, 0.0f);
}

// ---------------- K6: e_out = e_in + relu(bn(e_new)), in place over d_out e-region ----------------
__global__ __launch_bounds__(256) void k_eapply(
    float* __restrict__ eio, const float* __restrict__ ein,
    const float* __restrict__ sc, const float* __restrict__ sh) {
  const size_t t = (size_t)blockIdx.x * 256 + threadIdx.x;   // float4 index
  // pull the streams ~128KB ahead (speculative prefetch: OOB is silently dropped)
  __builtin_prefetch((const void*)(((const float4*)eio) + t + 8192), 0, 0);
  __builtin_prefetch((const void*)(((const float4*)ein) + t + 8192), 0, 0);
  float4 x = ((const float4*)eio)[t];
  float4 a = ((const float4*)ein)[t];
  const int nb = ((int)t & 31) * 4;
  float4 o;
  o.x = a.x + fmaxf(x.x * sc[nb + 0] + sh[nb + 0], 0.0f);
  o.y = a.y + fmaxf(x.y * sc[nb + 1] + sh[nb + 1], 0.0f);
  o.z = a.z + fmaxf(x.z * sc[nb + 2] + sh[nb + 2], 0.0f);
  o.w = a.w + fmaxf(x.w * sc[nb + 3] + sh[nb + 3], 0.0f);
  ((float4*)eio)[t] = o;
}

extern "C" void kernel_launch(void* const* d_in, const int* in_sizes, int n_in,
                              void* d_out, int out_size, void* d_ws, size_t ws_size,
                              hipStream_t stream) {
  const float* h_in = (const float*)d_in[0];
  const float* e_in = (const float*)d_in[1];
  // d_in[2] = graph (all ones, unused by the reference math)
  const float* Uw = (const float*)d_in[3];  const float* Ub = (const float*)d_in[4];
  const float* Vw = (const float*)d_in[5];  const float* Vb = (const float*)d_in[6];
  const float* Aw = (const float*)d_in[7];  const float* Ab = (const float*)d_in[8];
  const float* Bw = (const float*)d_in[9];  const float* Bb = (const float*)d_in[10];
  const float* Cw = (const float*)d_in[11]; const float* Cb = (const float*)d_in[12];
  const float* gamma_h = (const float*)d_in[13]; const float* beta_h = (const float*)d_in[14];
  const float* gamma_e = (const float*)d_in[15]; const float* beta_e = (const float*)d_in[16];

  float* out   = (float*)d_out;
  float* h_out = out;                 // B*V*H
  float* e_out = out + BVH;           // B*V*V*H ; also holds e_new between K2 and K6

  float* ws    = (float*)d_ws;
  float* Uh    = ws;
  float* Vh    = ws + 1 * BVH;
  float* Ah    = ws + 2 * BVH;
  float* Bh    = ws + 3 * BVH;
  float* agg   = ws + 4 * BVH;
  float* h_new = ws + 5 * BVH;
  float* st    = ws + 6 * BVH;        // stats block
  float* sum_e   = st;        float* sumsq_e = st + 128;
  float* sum_h   = st + 256;  float* sumsq_h = st + 384;
  float* sc_e    = st + 512;  float* sh_e    = st + 640;
  float* sc_h    = st + 768;  float* sh_h    = st + 896;

  k_zero<<<1, 256, 0, stream>>>(sum_e);  // zeros sum_e + sumsq_e (contiguous 256 floats)

  k_hlin<<<256, 256, 0, stream>>>(h_in, Uw, Ub, Vw, Vb, Aw, Ab, Bw, Bb, Uh, Vh, Ah, Bh);

  k_emain<<<Bn * Vn, 256, 0, stream>>>(e_in, Cw, Cb, Ah, Bh, Vh, e_out, agg, sum_e, sumsq_e);

  k_hstats<<<Hd, 256, 0, stream>>>(Uh, agg, h_new, sum_h, sumsq_h);

  k_fin<<<1, 128, 0, stream>>>(sum_e, sumsq_e, sum_h, sumsq_h,
                               gamma_h, beta_h, gamma_e, beta_e,
                               sc_e, sh_e, sc_h, sh_h);

  k_happly<<<BVH / 256, 256, 0, stream>>>(h_new, sc_h, sh_h, h_in, h_out);

  const int e4 = (Bn * Vn * Vn * Hd) / 4;          // float4 count
  k_eapply<<<e4 / 256, 256, 0, stream>>>(e_out, e_in, sc_e, sh_e);
}